// PriorEncoder_78718160601170
// MI455X (gfx1250) — compile-verified
//
#include <hip/hip_runtime.h>

// PriorEncoder embedding gather:
//   mean[n, d] = Wm[d, idx[n]]
//   var [n, d] = exp(2 * Wl[d, idx[n]])
// D = 64 fixed. Pure memory-bound gather; WMMA not applicable (0.008 FLOP/B).
// Roofline: ~256MB compulsory weight reads (cached in 192MB L2, ~2x reuse)
//           + 512MB NT output stream => ~33us floor at 23.3 TB/s.
//
// Mapping: 16 lanes per index, 4 consecutive d's per lane.
//   - all offsets fit in u32 -> uniform SGPR base + 32-bit VGPR offset
//     (gfx1250 GVS addressing), minimal VALU address setup
//   - weight reads: RT (default) so the weight set stays L2-resident
//   - output writes: non-temporal b128 stores (th:TH_STORE_NT) so the
//     512MB output stream does not evict the weights from L2

typedef __attribute__((ext_vector_type(4))) float v4f;

__global__ __launch_bounds__(256) void prior_encoder_gather_kernel(
    const int* __restrict__ idx,
    const float* __restrict__ Wm,   // [64, vocab] row-major
    const float* __restrict__ Wl,   // [64, vocab] row-major
    float* __restrict__ out,        // [N*64 mean | N*64 var]
    unsigned N, unsigned vocab)
{
    unsigned t = blockIdx.x * blockDim.x + threadIdx.x;  // < 16M, 32-bit safe
    unsigned n = t >> 4;                                 // 16 threads per index
    if (n >= N) return;
    unsigned dbase = (t & 15u) << 2;                     // d = dbase .. dbase+3

    // Keep the index stream ahead of the dependent gathers.
    if ((threadIdx.x & 63u) == 0u) {
        __builtin_prefetch(idx + n + 4096, 0, 3);        // -> global_prefetch_b8
    }

    unsigned col = (unsigned)idx[n];                     // broadcast per 16 lanes

    // u32 element offsets off a uniform base -> saddr-form global loads.
    unsigned off = dbase * vocab + col;                  // < 32M elements
    float m0 = Wm[off];
    float m1 = Wm[off + vocab];
    float m2 = Wm[off + 2u * vocab];
    float m3 = Wm[off + 3u * vocab];
    float l0 = Wl[off];
    float l1 = Wl[off + vocab];
    float l2 = Wl[off + 2u * vocab];
    float l3 = Wl[off + 3u * vocab];

    v4f mv, vv;
    mv.x = m0; mv.y = m1; mv.z = m2; mv.w = m3;
    vv.x = __expf(l0 + l0);                              // exp(2*lv) -> v_exp_f32
    vv.y = __expf(l1 + l1);
    vv.z = __expf(l2 + l2);
    vv.w = __expf(l3 + l3);

    unsigned obase = (n << 6) + dbase;                   // < 64M elements
    // NT 16B stores: wave writes 512B contiguous per matrix, L2-bypassing.
    __builtin_nontemporal_store(mv, (v4f*)(out + obase));
    __builtin_nontemporal_store(vv, (v4f*)(out + (N << 6) + obase));
}

extern "C" void kernel_launch(void* const* d_in, const int* in_sizes, int n_in,
                              void* d_out, int out_size, void* d_ws, size_t ws_size,
                              hipStream_t stream)
{
    const int*   idx = (const int*)d_in[0];
    const float* Wm  = (const float*)d_in[1];
    const float* Wl  = (const float*)d_in[2];
    float*       out = (float*)d_out;

    const unsigned N     = (unsigned)in_sizes[0];
    const unsigned vocab = (unsigned)in_sizes[1] / 64u;  // weight is [64, vocab]

    const unsigned long long threads = (unsigned long long)N * 16ull;
    const int block = 256;
    const unsigned grid = (unsigned)((threads + block - 1) / block);

    prior_encoder_gather_kernel<<<grid, block, 0, stream>>>(
        idx, Wm, Wl, out, N, vocab);
}